// BayesianAtlas_76596446757149
// MI455X (gfx1250) — compile-verified
//
#include <hip/hip_runtime.h>

// ===========================================================================
// BayesianAtlas (LDDMM Hamiltonian shooting), MI455X / gfx1250.
//
// Phase 1 (shoot_kernel): 16 blocks (1/batch), 6 sequential symplectic steps
//   on 64 control points; analytic MLP forward+backward per ordered pair;
//   writes the 7 (p_t,q_t) states to d_ws.
// Phase 2 (flow_kernel): 256 blocks; each wave owns 16 template points and
//   runs the pairwise-kernel MLP as f32 WMMA 16x16x4 tiles (K=4 exactly for
//   layer 1; K=10 -> 3 zero-padded K=4 chunks for layers 2/3). The two MLP
//   input orderings run as two independent WMMA pipelines (separate LDS
//   staging tiles) so the scheduler can interleave them.
// All transcendentals are branch-free (v_exp_f32 + v_rcp_f32) to keep the
// inner loop free of libm EXEC-mask churn.
// ===========================================================================

typedef float v2f __attribute__((ext_vector_type(2)));
typedef float v8f __attribute__((ext_vector_type(8)));

#define BTS    16
#define NCP    64
#define NTPL   2048
#define HID    10
#define TSTEPS 6
#define TP1    7
#define DT     0.2f
#define STRD   20          // staging tile row stride (floats): pad 10 -> 20

// Branch-free tanh: (e^{2x}-1)/(e^{2x}+1), clamped so e^{2x} stays finite.
__device__ __forceinline__ float fast_tanh(float x)
{
    const float t = fminf(fmaxf(x, -15.f), 15.f);
    const float e = __expf(2.f * t);                    // v_exp_f32 path
    return (e - 1.f) * __builtin_amdgcn_rcpf(e + 1.f);  // v_rcp_f32
}

// ---------------------------------------------------------------------------
// Scalar MLP forward / backward (used in the sequential shooting phase).
// ---------------------------------------------------------------------------
__device__ __forceinline__ void mlp_fwd(const float* __restrict__ sW1,
                                        const float* __restrict__ sb1,
                                        const float* __restrict__ sW2,
                                        const float* __restrict__ sb2,
                                        const float* __restrict__ sW3,
                                        const float* __restrict__ sb3,
                                        float z0, float z1, float z2, float z3,
                                        float* h1, float* h2, float* o)
{
#pragma unroll
    for (int n = 0; n < HID; ++n)
        h1[n] = fast_tanh(z0 * sW1[0 * HID + n] + z1 * sW1[1 * HID + n] +
                          z2 * sW1[2 * HID + n] + z3 * sW1[3 * HID + n] + sb1[n]);
#pragma unroll
    for (int n = 0; n < HID; ++n) {
        float t = sb2[n];
#pragma unroll
        for (int k = 0; k < HID; ++k) t += h1[k] * sW2[k * HID + n];
        h2[n] = fast_tanh(t);
    }
#pragma unroll
    for (int n = 0; n < 3; ++n) {
        float t = sb3[n];
#pragma unroll
        for (int k = 0; k < HID; ++k) t += h2[k] * sW3[k * 3 + n];
        o[n] = t;
    }
}

// Backward: given cotangent w[3] on the outputs, return gz = dL/dz (4-vec).
__device__ __forceinline__ void mlp_bwd(const float* __restrict__ sW1,
                                        const float* __restrict__ sW2,
                                        const float* __restrict__ sW3,
                                        const float* h1, const float* h2,
                                        float w0, float w1, float w2,
                                        float* gz)
{
    float g2[HID], g1[HID];
#pragma unroll
    for (int k = 0; k < HID; ++k) {
        float s = w0 * sW3[k * 3 + 0] + w1 * sW3[k * 3 + 1] + w2 * sW3[k * 3 + 2];
        g2[k] = s * (1.f - h2[k] * h2[k]);
    }
#pragma unroll
    for (int k = 0; k < HID; ++k) {
        float s = 0.f;
#pragma unroll
        for (int nn = 0; nn < HID; ++nn) s += g2[nn] * sW2[k * HID + nn];
        g1[k] = s * (1.f - h1[k] * h1[k]);
    }
#pragma unroll
    for (int d = 0; d < 4; ++d) {
        float s = 0.f;
#pragma unroll
        for (int nn = 0; nn < HID; ++nn) s += g1[nn] * sW1[d * HID + nn];
        gz[d] = s;
    }
}

// ---------------------------------------------------------------------------
// Phase 1: Hamiltonian shooting. H = 0.5 * sum_ij q_i^T S(f(p_i,p_j)) q_j
// (the symmetrized kernel collapses to this because S is symmetric).
// p' = p + DT * dH/dq, q' = q - DT * dH/dp.
// Thread t handles point i = t/4, j = (t%4) + 4*jj; each ordered pair is
// evaluated with BOTH input orders so all contributions land on point i
// (no LDS atomics, at 2x MLP cost on the small phase).
// ---------------------------------------------------------------------------
__global__ __launch_bounds__(256) void shoot_kernel(
    const float* __restrict__ q0, const float* __restrict__ cp,
    const float* __restrict__ W1, const float* __restrict__ b1,
    const float* __restrict__ W2, const float* __restrict__ b2,
    const float* __restrict__ W3, const float* __restrict__ b3,
    float* __restrict__ pT, float* __restrict__ qT)
{
    __shared__ float sW1[40], sb1[HID], sW2[100], sb2[HID], sW3[30], sb3[3];
    __shared__ float sP[2 * NCP], sQ[2 * NCP];
    __shared__ float red[256 * 4];

    const int b = blockIdx.x, tid = threadIdx.x;
    if (tid < 40)  sW1[tid] = W1[tid];
    if (tid < 100) sW2[tid] = W2[tid];
    if (tid < 30)  sW3[tid] = W3[tid];
    if (tid < HID) { sb1[tid] = b1[tid]; sb2[tid] = b2[tid]; }
    if (tid < 3)   sb3[tid] = b3[tid];
    if (tid < 2 * NCP) { sP[tid] = cp[tid]; sQ[tid] = q0[b * 2 * NCP + tid]; }
    __syncthreads();
    if (tid < 2 * NCP) {             // state t = 0
        pT[b * TP1 * 2 * NCP + tid] = sP[tid];
        qT[b * TP1 * 2 * NCP + tid] = sQ[tid];
    }

    const int i = tid >> 2, sub = tid & 3;
#pragma unroll 1
    for (int t = 0; t < TSTEPS; ++t) {
        const float pix = sP[2 * i], piy = sP[2 * i + 1];
        const float qix = sQ[2 * i], qiy = sQ[2 * i + 1];
        float dqx = 0.f, dqy = 0.f, dpx = 0.f, dpy = 0.f;
#pragma unroll 1
        for (int jj = 0; jj < 16; ++jj) {
            const int j = (jj << 2) | sub;
            const float pjx = sP[2 * j], pjy = sP[2 * j + 1];
            const float qjx = sQ[2 * j], qjy = sQ[2 * j + 1];
            float h1A[HID], h2A[HID], oA[3], h1B[HID], h2B[HID], oB[3];
            float gzA[4], gzB[4];
            mlp_fwd(sW1, sb1, sW2, sb2, sW3, sb3, pix, piy, pjx, pjy, h1A, h2A, oA);
            mlp_fwd(sW1, sb1, sW2, sb2, sW3, sb3, pjx, pjy, pix, piy, h1B, h2B, oB);
            const float e1A = __expf(oA[1]), e2A = __expf(oA[2]);
            const float e1B = __expf(oB[1]), e2B = __expf(oB[2]);
            const float w0 = 0.5f * (qix * qjy + qiy * qjx);
            // pair (i,j): grad flows into first half of z = [p_i, p_j]
            mlp_bwd(sW1, sW2, sW3, h1A, h2A,
                    w0, 0.5f * e1A * qix * qjx, 0.5f * e2A * qiy * qjy, gzA);
            // pair (j,i): grad flows into second half of z = [p_j, p_i]
            mlp_bwd(sW1, sW2, sW3, h1B, h2B,
                    w0, 0.5f * e1B * qjx * qix, 0.5f * e2B * qjy * qiy, gzB);
            const float offs = oA[0] + oB[0];
            dqx += 0.5f * ((e1A + e1B) * qjx + offs * qjy);
            dqy += 0.5f * (offs * qjx + (e2A + e2B) * qjy);
            dpx += gzA[0] + gzB[2];
            dpy += gzA[1] + gzB[3];
        }
        red[tid * 4 + 0] = dqx; red[tid * 4 + 1] = dqy;
        red[tid * 4 + 2] = dpx; red[tid * 4 + 3] = dpy;
        __syncthreads();
        if (tid < NCP) {
            float Dqx = 0.f, Dqy = 0.f, Dpx = 0.f, Dpy = 0.f;
#pragma unroll
            for (int s = 0; s < 4; ++s) {
                const int base = (tid * 4 + s) * 4;
                Dqx += red[base + 0]; Dqy += red[base + 1];
                Dpx += red[base + 2]; Dpy += red[base + 3];
            }
            const float pnx = sP[2 * tid] + DT * Dqx;
            const float pny = sP[2 * tid + 1] + DT * Dqy;
            const float qnx = sQ[2 * tid] - DT * Dpx;
            const float qny = sQ[2 * tid + 1] - DT * Dpy;
            sP[2 * tid] = pnx; sP[2 * tid + 1] = pny;
            sQ[2 * tid] = qnx; sQ[2 * tid + 1] = qny;
            const int off = b * TP1 * 2 * NCP + (t + 1) * 2 * NCP + tid * 2;
            pT[off] = pnx; pT[off + 1] = pny;
            qT[off] = qnx; qT[off + 1] = qny;
        }
        __syncthreads();
    }
}

// ---------------------------------------------------------------------------
// Phase 2: flow 2048 template points through the 7 stored states.
// One wave owns 16 points; per (state t, control point j) both MLP input
// orderings run as independent v_wmma_f32_16x16x4_f32 pipelines.
//
// WMMA f32 16x16x4 VGPR layouts (ISA 7.12.2):
//   A (16x4):  lane l holds row m=l&15, float2 {z[k0], z[k0+1]}, k0=(l<16?0:2)
//   B (4x16):  lane l holds col n=l&15, float2 {W[k0][n], W[k0+1][n]}
//   C/D:       VGPR r, lane l -> row (r + 8*(l>=16)), col l&15
// Staging tiles use row stride 20 floats: unconditional 32-lane stores
// (cols 10..15 store tanh(0)=0 = exactly the K-padding layers 2/3 need),
// conflict-free on both store halves and b64 A-chunk reloads.
// ---------------------------------------------------------------------------
__global__ __launch_bounds__(256) void flow_kernel(
    const float* __restrict__ tpl,
    const float* __restrict__ W1, const float* __restrict__ b1,
    const float* __restrict__ W2, const float* __restrict__ b2,
    const float* __restrict__ W3, const float* __restrict__ b3,
    const float* __restrict__ pT, const float* __restrict__ qT,
    float* __restrict__ out)
{
    __shared__ float sP[TP1 * 2 * NCP];                  // this batch's p_t
    __shared__ float sQ[TP1 * 2 * NCP];                  // this batch's q_t
    __shared__ __align__(16) float sX[8 * 32];           // 8 waves x 16 pts x 2
    __shared__ __align__(16) float stg[8][4][16 * STRD]; // {1A,1B,2A,2B}/wave
    __shared__ __align__(16) float obf[8][2][16 * 4];    // MLP outputs A/B

    const int b = blockIdx.y, tile = blockIdx.x, tid = threadIdx.x;
    const int wave = tid >> 5, lane = tid & 31;
    const int m = lane & 15, hi = lane >> 4;
    const int n = m;                 // B/C/D column owned by this lane
    const int k0 = hi ? 2 : 0;       // K pair owned by this lane

    for (int idx = tid; idx < TP1 * 2 * NCP; idx += 256) {
        sP[idx] = pT[b * TP1 * 2 * NCP + idx];
        sQ[idx] = qT[b * TP1 * 2 * NCP + idx];
    }
    const int pt = tile * 128 + wave * 16 + m;
    if (hi == 0) {
        sX[wave * 32 + 2 * m + 0] = tpl[pt * 2 + 0];
        sX[wave * 32 + 2 * m + 1] = tpl[pt * 2 + 1];
    }

    // Zero-padded B operands + per-lane bias scalars (resident in VGPRs).
    v2f B1op, B2op[3], B3op[3];
    B1op.x = (n < HID) ? W1[(k0 + 0) * HID + n] : 0.f;
    B1op.y = (n < HID) ? W1[(k0 + 1) * HID + n] : 0.f;
#pragma unroll
    for (int c = 0; c < 3; ++c) {
        const int k = 4 * c + k0;
        B2op[c].x = (n < HID && k < HID)       ? W2[k * HID + n]       : 0.f;
        B2op[c].y = (n < HID && (k + 1) < HID) ? W2[(k + 1) * HID + n] : 0.f;
        B3op[c].x = (n < 3 && k < HID)         ? W3[k * 3 + n]         : 0.f;
        B3op[c].y = (n < 3 && (k + 1) < HID)   ? W3[(k + 1) * 3 + n]   : 0.f;
    }
    const float bb1 = (n < HID) ? b1[n] : 0.f;
    const float bb2 = (n < HID) ? b2[n] : 0.f;
    const float bb3 = (n < 3)   ? b3[n] : 0.f;
    __syncthreads();

    float* const st1A = &stg[wave][0][0];
    float* const st1B = &stg[wave][1][0];
    float* const st2A = &stg[wave][2][0];
    float* const st2B = &stg[wave][3][0];
    float* const oAb  = &obf[wave][0][0];
    float* const oBb  = &obf[wave][1][0];
    const int srow = (0 + 8 * hi) * STRD + n;   // store base (row offset added per r)
    const int lrow = m * STRD + 2 * hi;         // A-chunk load base

    const v8f vzero = {0.f, 0.f, 0.f, 0.f, 0.f, 0.f, 0.f, 0.f};

#pragma unroll 1
    for (int t = 0; t < TP1; ++t) {
        const float xmx = sX[wave * 32 + 2 * m + 0];
        const float xmy = sX[wave * 32 + 2 * m + 1];
        float acc = 0.f;
#pragma unroll 1
        for (int j = 0; j < NCP; ++j) {
            const float pjx = sP[(t * NCP + j) * 2 + 0];
            const float pjy = sP[(t * NCP + j) * 2 + 1];
            const float qjx = sQ[(t * NCP + j) * 2 + 0];
            const float qjy = sQ[(t * NCP + j) * 2 + 1];

            // A row m: dirA z=[x_m, p_j], dirB z=[p_j, x_m]
            v2f a1A, a1B;
            a1A.x = hi ? pjx : xmx;  a1A.y = hi ? pjy : xmy;
            a1B.x = hi ? xmx : pjx;  a1B.y = hi ? xmy : pjy;

            // ---- layer 1 (K = 4, exact), both directions ----
            v8f c1A = __builtin_amdgcn_wmma_f32_16x16x4_f32(
                false, a1A, false, B1op, (short)0, vzero, false, false);
            v8f c1B = __builtin_amdgcn_wmma_f32_16x16x4_f32(
                false, a1B, false, B1op, (short)0, vzero, false, false);
#pragma unroll
            for (int r = 0; r < 8; ++r) {
                st1A[srow + r * STRD] = fast_tanh(c1A[r] + bb1);
                st1B[srow + r * STRD] = fast_tanh(c1B[r] + bb1);
            }
            __builtin_amdgcn_wave_barrier();

            // ---- layer 2 (K = 10 -> 3 padded K=4 chunks) ----
            const v2f a2A0 = *(const v2f*)(st1A + lrow + 0);
            const v2f a2A1 = *(const v2f*)(st1A + lrow + 4);
            const v2f a2A2 = *(const v2f*)(st1A + lrow + 8);
            const v2f a2B0 = *(const v2f*)(st1B + lrow + 0);
            const v2f a2B1 = *(const v2f*)(st1B + lrow + 4);
            const v2f a2B2 = *(const v2f*)(st1B + lrow + 8);
            v8f c2A = __builtin_amdgcn_wmma_f32_16x16x4_f32(
                false, a2A0, false, B2op[0], (short)0, vzero, false, false);
            v8f c2B = __builtin_amdgcn_wmma_f32_16x16x4_f32(
                false, a2B0, false, B2op[0], (short)0, vzero, false, false);
            c2A = __builtin_amdgcn_wmma_f32_16x16x4_f32(
                false, a2A1, false, B2op[1], (short)0, c2A, false, false);
            c2B = __builtin_amdgcn_wmma_f32_16x16x4_f32(
                false, a2B1, false, B2op[1], (short)0, c2B, false, false);
            c2A = __builtin_amdgcn_wmma_f32_16x16x4_f32(
                false, a2A2, false, B2op[2], (short)0, c2A, false, false);
            c2B = __builtin_amdgcn_wmma_f32_16x16x4_f32(
                false, a2B2, false, B2op[2], (short)0, c2B, false, false);
#pragma unroll
            for (int r = 0; r < 8; ++r) {
                st2A[srow + r * STRD] = fast_tanh(c2A[r] + bb2);
                st2B[srow + r * STRD] = fast_tanh(c2B[r] + bb2);
            }
            __builtin_amdgcn_wave_barrier();

            // ---- layer 3 (K = 10 -> 3 padded K=4 chunks, N = 3) ----
            const v2f a3A0 = *(const v2f*)(st2A + lrow + 0);
            const v2f a3A1 = *(const v2f*)(st2A + lrow + 4);
            const v2f a3A2 = *(const v2f*)(st2A + lrow + 8);
            const v2f a3B0 = *(const v2f*)(st2B + lrow + 0);
            const v2f a3B1 = *(const v2f*)(st2B + lrow + 4);
            const v2f a3B2 = *(const v2f*)(st2B + lrow + 8);
            v8f c3A = __builtin_amdgcn_wmma_f32_16x16x4_f32(
                false, a3A0, false, B3op[0], (short)0, vzero, false, false);
            v8f c3B = __builtin_amdgcn_wmma_f32_16x16x4_f32(
                false, a3B0, false, B3op[0], (short)0, vzero, false, false);
            c3A = __builtin_amdgcn_wmma_f32_16x16x4_f32(
                false, a3A1, false, B3op[1], (short)0, c3A, false, false);
            c3B = __builtin_amdgcn_wmma_f32_16x16x4_f32(
                false, a3B1, false, B3op[1], (short)0, c3B, false, false);
            c3A = __builtin_amdgcn_wmma_f32_16x16x4_f32(
                false, a3A2, false, B3op[2], (short)0, c3A, false, false);
            c3B = __builtin_amdgcn_wmma_f32_16x16x4_f32(
                false, a3B2, false, B3op[2], (short)0, c3B, false, false);
            if (n < 3) {
#pragma unroll
                for (int r = 0; r < 8; ++r) {
                    oAb[(r + 8 * hi) * 4 + n] = c3A[r] + bb3;
                    oBb[(r + 8 * hi) * 4 + n] = c3B[r] + bb3;
                }
            }
            __builtin_amdgcn_wave_barrier();

            // K_mj = 0.5*(S(o_A)+S(o_B)); lanes<16 accumulate vel.x, >=16 vel.y
            const float o0A = oAb[m * 4 + 0], o1A = oAb[m * 4 + 1], o2A = oAb[m * 4 + 2];
            const float o0B = oBb[m * 4 + 0], o1B = oBb[m * 4 + 1], o2B = oBb[m * 4 + 2];
            const float offk = 0.5f * (o0A + o0B);
            const float e1 = 0.5f * (__expf(o1A) + __expf(o1B));
            const float e2 = 0.5f * (__expf(o2A) + __expf(o2B));
            acc += hi ? (offk * qjx + e2 * qjy) : (e1 * qjx + offk * qjy);
            __builtin_amdgcn_wave_barrier();
        }
        // x <- x + DT * vel (each lane owns one (point, component))
        sX[wave * 32 + 2 * m + hi] = (hi ? xmy : xmx) + DT * acc;
        __builtin_amdgcn_wave_barrier();
    }
    out[(b * NTPL + pt) * 2 + hi] = sX[wave * 32 + 2 * m + hi];
}

// ---------------------------------------------------------------------------
extern "C" void kernel_launch(void* const* d_in, const int* in_sizes, int n_in,
                              void* d_out, int out_size, void* d_ws, size_t ws_size,
                              hipStream_t stream)
{
    (void)in_sizes; (void)n_in; (void)out_size; (void)ws_size;
    const float* q0  = (const float*)d_in[0];
    const float* tpl = (const float*)d_in[1];
    const float* cp  = (const float*)d_in[2];
    const float* W1  = (const float*)d_in[3];
    const float* b1  = (const float*)d_in[4];
    const float* W2  = (const float*)d_in[5];
    const float* b2  = (const float*)d_in[6];
    const float* W3  = (const float*)d_in[7];
    const float* b3  = (const float*)d_in[8];
    float* out = (float*)d_out;

    float* pT = (float*)d_ws;                      // [16][7][64][2]
    float* qT = pT + BTS * TP1 * NCP * 2;          // [16][7][64][2]

    shoot_kernel<<<BTS, 256, 0, stream>>>(q0, cp, W1, b1, W2, b2, W3, b3, pT, qT);
    flow_kernel<<<dim3(NTPL / 128, BTS), 256, 0, stream>>>(
        tpl, W1, b1, W2, b2, W3, b3, pT, qT, out);
}